// IS_FTT_1_layer_relu_61512521613620
// MI455X (gfx1250) — compile-verified
//
#include <hip/hip_runtime.h>
#include <math.h>

// ---------------------------------------------------------------------------
// TT-network fused via TT-matrix composition (OUT_MODES_1 == IN_MODES_2 =>
// W1@W2 is a TT matrix with rank 3*3 = 9). Chain per batch row:
//   3072 -> (1024,9) -> (256,18) -> (64,45) -> (80,9) -> 10 -> log_softmax
// Combined cores stored in d_ws as TRANSPOSED, ZERO-PADDED B-matrices
// BT[col][Kpad] so WMMA B fragments are single unconditional ds_load_b64:
//   B0T: 16 x  4   (C=9,  K=3 ; step0)
//   B1T: 32 x 36   (C=18, K=36; step1)
//   B2T: 48 x 72   (C=45, K=72; step2)
//   B3T: 16 x 36   (C=9,  K=36; step3)
//   G4 :      72   (C=1,  K=72; step4, scalar)
// Every intermediate is written directly in the NEXT step's A-row-major
// layout A[R][Kpad], so A fragments are also single ds_load_b64 with no
// div/mod in the K loop. Steps 0-3 use V_WMMA_F32_16X16X4_F32 (fp32 math,
// matching the reference dtype; K=4 fits the tiny rank-contraction dims).
// ---------------------------------------------------------------------------

typedef __attribute__((ext_vector_type(2))) float v2f;
typedef __attribute__((ext_vector_type(8))) float v8f;

#define B0T_OFF 0      // 16*4   =   64
#define B1T_OFF 64     // 32*36  = 1152
#define B2T_OFF 1216   // 48*72  = 3456
#define B3T_OFF 4672   // 16*36  =  576
#define G4_OFF  5248   // 72
#define G_TOTAL 5320

// ------------------------- kernel 1: combine cores -------------------------
// G_k[(r,s), m, p, (r',s')] = sum_n c1_k[r,m,n,r'] * c2_k[s,n,p,s']
// stored transposed+padded: BT[c][kk], kk = m*9 + (r*3+s), c = p*9 + (r'*3+s')
__global__ void __launch_bounds__(256)
tt_combine_cores(const float* __restrict__ c1_0, const float* __restrict__ c1_1,
                 const float* __restrict__ c1_2, const float* __restrict__ c1_3,
                 const float* __restrict__ c1_4,
                 const float* __restrict__ c2_0, const float* __restrict__ c2_1,
                 const float* __restrict__ c2_2, const float* __restrict__ c2_3,
                 const float* __restrict__ c2_4,
                 float* __restrict__ G)
{
  const int tid = threadIdx.x;
  const int stride = blockDim.x;

  // B0T: 16 x 4 ; value[kk=m][c=rp*3+sp] = sum_n c1_0[0,m,n,rp]*c2_0[0,n,0,sp]
  for (int i = tid; i < 64; i += stride) {
    int c = i >> 2, kk = i & 3;
    float acc = 0.f;
    if (c < 9 && kk < 3) {
      int rp = c / 3, sp = c % 3;
      for (int n = 0; n < 10; ++n)
        acc += c1_0[(kk * 10 + n) * 3 + rp] * c2_0[n * 3 + sp];
    }
    G[B0T_OFF + i] = acc;
  }
  // B1T: 32 x 36
  for (int i = tid; i < 1152; i += stride) {
    int c = i / 36, kk = i % 36;
    float acc = 0.f;
    if (c < 18) {
      int m = kk / 9, t = kk % 9, r = t / 3, s = t % 3;
      int p = c / 9, tp = c % 9, rp = tp / 3, sp = tp % 3;
      for (int n = 0; n < 20; ++n)
        acc += c1_1[((r * 4 + m) * 20 + n) * 3 + rp] *
               c2_1[((s * 20 + n) * 2 + p) * 3 + sp];
    }
    G[B1T_OFF + i] = acc;
  }
  // B2T: 48 x 72
  for (int i = tid; i < 3456; i += stride) {
    int c = i / 72, kk = i % 72;
    float acc = 0.f;
    if (c < 45) {
      int m = kk / 9, t = kk % 9, r = t / 3, s = t % 3;
      int p = c / 9, tp = c % 9, rp = tp / 3, sp = tp % 3;
      for (int n = 0; n < 50; ++n)
        acc += c1_2[((r * 8 + m) * 50 + n) * 3 + rp] *
               c2_2[((s * 50 + n) * 5 + p) * 3 + sp];
    }
    G[B2T_OFF + i] = acc;
  }
  // B3T: 16 x 36  (p == 0)
  for (int i = tid; i < 576; i += stride) {
    int c = i / 36, kk = i % 36;
    float acc = 0.f;
    if (c < 9) {
      int m = kk / 9, t = kk % 9, r = t / 3, s = t % 3;
      int rp = c / 3, sp = c % 3;
      for (int n = 0; n < 20; ++n)
        acc += c1_3[((r * 4 + m) * 20 + n) * 3 + rp] *
               c2_3[(s * 20 + n) * 3 + sp];
    }
    G[B3T_OFF + i] = acc;
  }
  // G4: 72  (K-vector, C = 1)
  for (int i = tid; i < 72; i += stride) {
    int m = i / 9, t = i % 9, r = t / 3, s = t % 3;
    float acc = 0.f;
    for (int n = 0; n < 10; ++n)
      acc += c1_4[(r * 8 + m) * 10 + n] * c2_4[s * 10 + n];
    G[G4_OFF + i] = acc;
  }
}

// --------------------- WMMA tile step (f32 16x16x4) ------------------------
// A[R][Kpad] row-major in LDS (K-padding pre-zeroed by the producer),
// BT[col][Kpad] in LDS (K and C padding zeroed). Both fragment loads are a
// single unconditional 8B-aligned ds_load_b64; no div/mod in the K loop.
// ScatterF maps (row, col) -> flat index in the NEXT step's A layout.
template <typename ScatterF>
__device__ __forceinline__ void
tt_gemm_step(const float* __restrict__ in, float* __restrict__ outBuf,
             const float* __restrict__ Bt,
             int Mtiles, int Ctiles, int Ktiles, int Kpad, int C,
             int lane, int waveId, ScatterF scat)
{
  const int half  = lane >> 4;    // 0 or 1 (ISA half-wave K split)
  const int l16   = lane & 15;
  const int khalf = half * 2;     // K offset of the .x element

  for (int tile = waveId; tile < Mtiles * Ctiles; tile += 4) {
    const int mt = tile / Ctiles, ct = tile - mt * Ctiles;
    const int M   = (mt << 4) + l16;   // A row for this lane
    const int col = (ct << 4) + l16;   // B/C column for this lane
    const float* arow = in + M * Kpad + khalf;
    const float* brow = Bt + col * Kpad + khalf;
    v8f acc = {0.f, 0.f, 0.f, 0.f, 0.f, 0.f, 0.f, 0.f};
#pragma unroll
    for (int kt = 0; kt < Ktiles; ++kt) {
      v2f a = *(const v2f*)(arow + (kt << 2));   // ds_load_b64
      v2f b = *(const v2f*)(brow + (kt << 2));   // ds_load_b64
      acc = __builtin_amdgcn_wmma_f32_16x16x4_f32(
          /*neg_a=*/false, a, /*neg_b=*/false, b,
          /*c_mod=*/(short)0, acc, /*reuse_a=*/false, /*reuse_b=*/false);
    }
    // C/D layout: VGPR j holds row (mt*16 + j + 8*half), col = lane&15.
    if (col < C) {
      const int rbase = (mt << 4) + half * 8;
#pragma unroll
      for (int j = 0; j < 8; ++j)
        outBuf[scat(rbase + j, col)] = acc[j];
    }
  }
}

// ----------------------- kernel 2: per-batch TT chain ----------------------
__global__ void __launch_bounds__(128)
tt_chain_kernel(const float* __restrict__ x, const float* __restrict__ G,
                float* __restrict__ out)
{
  extern __shared__ float smem[];
  float* Gs   = smem;                  // 5320 floats (padded/transposed cores)
  float* buf0 = smem + G_TOTAL;        // 4608 floats (A0:4096, A2:4608, A4:720)
  float* buf1 = buf0 + 4608;           // 9216 floats (A1:9216, A3:2880)

  const int tid = threadIdx.x;
  const int lane = tid & 31;
  const int waveId = tid >> 5;
  const int b = blockIdx.x;            // one workgroup per batch row

  for (int i = tid; i < G_TOTAL; i += 128) Gs[i] = G[i];
  // A0[R][4]: A0[R*4+kk] = x_pad[kk*1024 + R]; nonzero only kk==0, R<177.
  for (int i = tid; i < 4096; i += 128) {
    int R = i >> 2, kk = i & 3;
    buf0[i] = (kk == 0 && R < 177) ? x[b * 177 + R] : 0.f;
  }
  __syncthreads();

  // step0: rows 1024, K 3->4, C 9 ; out -> A1[xc][36], xc=R&255, m=R>>8
  tt_gemm_step(buf0, buf1, Gs + B0T_OFF, 64, 1, 1, 4, 9, lane, waveId,
               [](int r, int c) { return (r & 255) * 36 + (r >> 8) * 9 + c; });
  __syncthreads();
  // step1: rows 256, K 36, C 18 ; out -> A2[(xc*2+n)][72], xc=R&31, m=R>>5
  tt_gemm_step(buf1, buf0, Gs + B1T_OFF, 16, 2, 9, 36, 18, lane, waveId,
               [](int r, int c) {
                 int n = c / 9, s = c - n * 9;
                 return ((r & 31) * 2 + n) * 72 + (r >> 5) * 9 + s;
               });
  __syncthreads();
  // step2: rows 64, K 72, C 45 ; rows R=(x*2+N) ; out -> A3[(xc*10+N*5+n)][36]
  tt_gemm_step(buf0, buf1, Gs + B2T_OFF, 4, 3, 18, 72, 45, lane, waveId,
               [](int r, int c) {
                 int n = c / 9, s = c - n * 9;
                 int xx = r >> 1, N = r & 1;
                 return ((xx & 7) * 10 + N * 5 + n) * 36 + (xx >> 3) * 9 + s;
               });
  __syncthreads();
  // step3: rows 80, K 36, C 9 ; rows R=(x*10+N) ; out -> A4[N][72]
  tt_gemm_step(buf1, buf0, Gs + B3T_OFF, 5, 1, 9, 36, 9, lane, waveId,
               [](int r, int c) {
                 int xx = r / 10, N = r - xx * 10;
                 return N * 72 + xx * 9 + c;
               });
  __syncthreads();

  // step4: logits[N] = sum_kk A4[N][kk] * G4[kk]   (K=72, C=1 -> scalar)
  if (tid < 10) {
    float acc = 0.f;
    for (int kk = 0; kk < 72; ++kk)
      acc += buf0[tid * 72 + kk] * Gs[G4_OFF + kk];
    buf1[tid] = acc;
  }
  __syncthreads();
  if (tid == 0) {
    float mx = buf1[0];
    for (int n = 1; n < 10; ++n) mx = fmaxf(mx, buf1[n]);
    float se = 0.f;
    for (int n = 0; n < 10; ++n) se += expf(buf1[n] - mx);
    buf1[10] = mx + logf(se);
  }
  __syncthreads();
  if (tid < 10)
    out[b * 10 + tid] = buf1[tid] - buf1[10];   // log_softmax
}

// ------------------------------- launcher ----------------------------------
extern "C" void kernel_launch(void* const* d_in, const int* in_sizes, int n_in,
                              void* d_out, int out_size, void* d_ws, size_t ws_size,
                              hipStream_t stream)
{
  const float* x    = (const float*)d_in[0];
  const float* c1_0 = (const float*)d_in[1];
  const float* c1_1 = (const float*)d_in[2];
  const float* c1_2 = (const float*)d_in[3];
  const float* c1_3 = (const float*)d_in[4];
  const float* c1_4 = (const float*)d_in[5];
  const float* c2_0 = (const float*)d_in[6];
  const float* c2_1 = (const float*)d_in[7];
  const float* c2_2 = (const float*)d_in[8];
  const float* c2_3 = (const float*)d_in[9];
  const float* c2_4 = (const float*)d_in[10];
  float* G = (float*)d_ws;   // 5320 floats of scratch (fully rewritten per call)

  tt_combine_cores<<<1, 256, 0, stream>>>(c1_0, c1_1, c1_2, c1_3, c1_4,
                                          c2_0, c2_1, c2_2, c2_3, c2_4, G);

  const size_t shmem = (size_t)(G_TOTAL + 4608 + 9216) * sizeof(float); // ~76.6 KB
  tt_chain_kernel<<<64, 128, shmem, stream>>>(x, G, (float*)d_out);
}